// PerturbationAttention_34213709480217
// MI455X (gfx1250) — compile-verified
//
#include <hip/hip_runtime.h>

// ---------------------------------------------------------------------------
// PerturbationAttention for MI455X (gfx1250, wave32)
//   delta [B=2048, C=512, D=256] f32  (1 GiB -> bandwidth bound, ~46us floor)
//   phase1: global -> LDS via GLOBAL_LOAD_ASYNC_TO_LDS_B128 (ASYNCcnt path,
//           no VGPR staging), then sigma[b,c] = ||delta[b,c,:]|| via
//           V_WMMA_F32_16X16X4_F32 row-sums (A = squared fragment, B = ones),
//           plus global max via int-bit atomics (sigma >= 0).
//   phase2: att = softmax_c(tanh(1 - sigma/max)); zero k smallest per batch
//           with top_k(-att) tie-to-lower-index semantics.
// ---------------------------------------------------------------------------

typedef float v2f __attribute__((ext_vector_type(2)));
typedef float v8f __attribute__((ext_vector_type(8)));

#define C_DIM 512
#define D_DIM 256
#define ROWS_PER_BLOCK 128   // 8 waves * 16 rows
#define KTILE 64
#define LDS_STRIDE 68        // 64 data + 4 pad floats: row bank-stride 4 ->
                             // 16-row x 4-col WMMA fragment covers all 64 banks

__global__ void pa_init(int* gmax_bits) { *gmax_bits = 0; }

__global__ __launch_bounds__(256) void pa_phase1_sigma(
    const float* __restrict__ delta, float* __restrict__ sigma_out,
    int* __restrict__ gmax_bits, int total_rows)
{
  __shared__ float tile[ROWS_PER_BLOCK * LDS_STRIDE];
  __shared__ int s_maxbits;

  const int t    = threadIdx.x;
  const int wave = t >> 5;
  const int lane = t & 31;
  const int m16  = lane & 15;   // N column in C/D layout, M row in A layout
  const int hi   = lane >> 4;   // lane half selects K pair (A) / M+8 (C,D)
  const long rows_base = (long)blockIdx.x * ROWS_PER_BLOCK;

  if (t == 0) s_maxbits = 0;

  v8f acc = {};
  v2f bones = {1.0f, 1.0f};     // B = ones: D[m,n] = sum_k A[m,k]

  const int lrow0 = t >> 4;        // 0..15
  const int c4    = (t & 15) << 2; // 0..60

  for (int kc = 0; kc < D_DIM; kc += KTILE) {
    __syncthreads();
    // 128 rows x 64 cols tile, copied global -> LDS with the CDNA5 async
    // data mover: 16B per lane per op, no VGPR staging, tracked by ASYNCcnt.
    // GVS addressing: saddr = delta base (SGPR pair), vaddr = u32 byte offset.
#pragma unroll
    for (int p = 0; p < 8; ++p) {
      const int lr = lrow0 + p * 16;
      const unsigned goff =
          (unsigned)(((rows_base + lr) * (long)D_DIM + kc + c4) * sizeof(float));
      const unsigned loff = (unsigned)(size_t)&tile[lr * LDS_STRIDE + c4];
      asm volatile("global_load_async_to_lds_b128 %0, %1, %2"
                   :
                   : "v"(loff), "v"(goff), "s"(delta)
                   : "memory");
    }
    // Drain this wave's async copies before the block barrier so every wave
    // observes a complete tile after the barrier.
    asm volatile("s_wait_asynccnt 0" ::: "memory");
    __syncthreads();

    // A fragment (16x4 f32): lane(<16) VGPR0/1 = K=kk+0/1, lane(>=16) K=kk+2/3
    const int wrow = wave * 16 + m16;
#pragma unroll
    for (int kk = 0; kk < KTILE; kk += 4) {
      const float* src = &tile[wrow * LDS_STRIDE + kk + hi * 2];
      v2f a; a.x = src[0]; a.y = src[1];
      a.x *= a.x; a.y *= a.y;  // square elementwise (VALU, co-exec with XDL)
      acc = __builtin_amdgcn_wmma_f32_16x16x4_f32(
          false, a, false, bones, (short)0, acc, false, false);
    }
  }

  // C/D layout: VGPR i on lane L holds D[M = i + 8*hi, N = m16]; all N equal.
  // Lanes 0..7 emit rows 0..7, lanes 16..23 emit rows 8..15 of this wave tile.
  if (m16 < 8) {
    const float s2   = acc[m16];
    const int   mrow = m16 + hi * 8;
    const long  grow = rows_base + wave * 16 + mrow;
    const float sig  = sqrtf(s2);
    if (grow < total_rows) sigma_out[grow] = sig;
    atomicMax(&s_maxbits, __float_as_int(sig));  // sig >= 0: int-bit compare ok
  }
  __syncthreads();
  if (t == 0) atomicMax(gmax_bits, s_maxbits);
}

__global__ __launch_bounds__(512) void pa_phase2_softmax_topk(
    float* __restrict__ att_io, const int* __restrict__ kptr,
    const int* __restrict__ gmax_bits)
{
  __shared__ float s_att[C_DIM];
  __shared__ float s_red[C_DIM];
  const int b = blockIdx.x;
  const int c = threadIdx.x;

  const float gmax = __int_as_float(*gmax_bits);
  const float sig  = att_io[(long)b * C_DIM + c];
  const float a    = tanhf(1.0f - sig / gmax);
  s_att[c] = a;
  s_red[c] = a;
  __syncthreads();

  // max over the 512-row for stable softmax
  for (int s = C_DIM / 2; s > 0; s >>= 1) {
    if (c < s) s_red[c] = fmaxf(s_red[c], s_red[c + s]);
    __syncthreads();
  }
  const float mx = s_red[0];
  __syncthreads();

  const float e = __expf(a - mx);
  s_red[c] = e;
  __syncthreads();
  for (int s = C_DIM / 2; s > 0; s >>= 1) {
    if (c < s) s_red[c] += s_red[c + s];
    __syncthreads();
  }
  const float total = s_red[0];

  // rank with top_k(-att) tie semantics: element removed iff
  //   #{j: a_j < a_i} + #{j: a_j == a_i && j < i} < k
  int r = 0;
  for (int j = 0; j < C_DIM; ++j) {
    const float aj = s_att[j];
    r += ((aj < a) || (aj == a && j < c)) ? 1 : 0;
  }

  const int k = *kptr;
  att_io[(long)b * C_DIM + c] = (r >= k) ? (e / total) : 0.0f;
}

extern "C" void kernel_launch(void* const* d_in, const int* in_sizes, int n_in,
                              void* d_out, int out_size, void* d_ws, size_t ws_size,
                              hipStream_t stream) {
  const float* delta = (const float*)d_in[0];
  const int*   kptr  = (const int*)d_in[1];
  float* out = (float*)d_out;
  int* gmax_bits = (int*)d_ws;              // 4 bytes of scratch

  const int total_rows = in_sizes[0] / D_DIM;           // B*C
  const int blocks1    = (total_rows + ROWS_PER_BLOCK - 1) / ROWS_PER_BLOCK;
  const int B          = total_rows / C_DIM;

  pa_init<<<1, 1, 0, stream>>>(gmax_bits);
  pa_phase1_sigma<<<blocks1, 256, 0, stream>>>(delta, out, gmax_bits, total_rows);
  pa_phase2_softmax_topk<<<B, C_DIM, 0, stream>>>(out, kptr, gmax_bits);
}